// QNN_67293547594100
// MI455X (gfx1250) — compile-verified
//
#include <hip/hip_runtime.h>

// ---------------------------------------------------------------------------
// QNN closed form: q = prod_i cos(pi/2 + 2*x_i) = prod_i (-sin(2*x_i))
// out[r] = { scale*q, -scale*q }
// B = 4,194,304 rows x 8 floats. Pure HBM-bandwidth-bound (160 MiB @ 23.3TB/s
// ~= 7us). No GEMM structure -> WMMA inapplicable; the gfx1250-specific path
// used here is the async global->LDS data mover (ASYNCcnt / s_wait_asynccnt)
// with double buffering.
// ---------------------------------------------------------------------------

#define BLOCK 256u
#define GRID  4096u
#define ROWS  4u
#define NQ    8u
// GRID * BLOCK * ROWS == 4,194,304 rows exactly
static_assert(GRID * BLOCK * ROWS == 4194304u, "row coverage mismatch");

#if __has_builtin(__builtin_amdgcn_global_load_async_to_lds_b128)
#define USE_ASYNC_LDS 1
#else
#define USE_ASYNC_LDS 0
#endif

// Parameter types for the async-LDS builtin (from the compiler diagnostic):
//   (v4i addrspace(1)* src, v4i addrspace(3)* dst, imm int offset, imm int cpol)
typedef int v4i __attribute__((ext_vector_type(4)));
typedef __attribute__((address_space(1))) v4i* gvec_t;
typedef __attribute__((address_space(3))) v4i* lvec_t;

template <int N>
__device__ __forceinline__ void wait_asynccnt_le() {
#if __has_builtin(__builtin_amdgcn_s_wait_asynccnt)
    __builtin_amdgcn_s_wait_asynccnt(N);
#else
    asm volatile("s_wait_asynccnt %0" ::"i"(N) : "memory");
#endif
}

__device__ __forceinline__ float row_prod(const float* v) {
    float prod = 1.0f;
#pragma unroll
    for (unsigned i = 0; i < NQ; ++i) {
        // cos(pi/2 + 2x) == -sin(2x); __sinf -> native v_sin_f32 path
        prod *= -__sinf(2.0f * v[i]);
    }
    return prod;
}

__global__ __launch_bounds__(BLOCK) void qnn_expz_kernel(
    const float* __restrict__ x,
    const float* __restrict__ scale_ptr,
    float* __restrict__ out) {
    const float scale     = scale_ptr[0];
    const unsigned tid    = threadIdx.x;
    const unsigned row0   = blockIdx.x * BLOCK + tid;
    const unsigned stride = GRID * BLOCK;

#if USE_ASYNC_LDS
    // Double-buffered staging: 2 x (256 rows x 8 floats) = 16 KB of 320 KB/WGP.
    __shared__ float tile[2][BLOCK * NQ];

    // Prologue: stage k = 0 (each thread copies its 32-byte row as 2 x b128)
    {
        float* g = const_cast<float*>(x) + (size_t)row0 * NQ;
        __builtin_amdgcn_global_load_async_to_lds_b128(
            (gvec_t)g, (lvec_t)&tile[0][tid * NQ], 0, 0);
        __builtin_amdgcn_global_load_async_to_lds_b128(
            (gvec_t)(g + 4), (lvec_t)&tile[0][tid * NQ + 4], 0, 0);
    }

    unsigned buf = 0;
#pragma unroll
    for (unsigned k = 0; k < ROWS; ++k) {
        const unsigned row = row0 + k * stride;
        if (k + 1 < ROWS) {
            // Issue next stage before waiting on the current one.
            float* g = const_cast<float*>(x) + (size_t)(row + stride) * NQ;
            __builtin_amdgcn_global_load_async_to_lds_b128(
                (gvec_t)g, (lvec_t)&tile[buf ^ 1][tid * NQ], 0, 0);
            __builtin_amdgcn_global_load_async_to_lds_b128(
                (gvec_t)(g + 4), (lvec_t)&tile[buf ^ 1][tid * NQ + 4], 0, 0);
            // 2 async ops still outstanding (next stage); current stage done
            // (async loads complete in order).
            wait_asynccnt_le<2>();
        } else {
            wait_asynccnt_le<0>();
        }
        const float q = scale * row_prod(&tile[buf][tid * NQ]);
        *reinterpret_cast<float2*>(out + (size_t)2 * row) = make_float2(q, -q);
        buf ^= 1;
    }
#else
    // Fallback: straight coalesced float4 loads (still bandwidth-optimal).
#pragma unroll
    for (unsigned k = 0; k < ROWS; ++k) {
        const unsigned row = row0 + k * stride;
        const float4* xv = reinterpret_cast<const float4*>(x + (size_t)row * NQ);
        const float4 a = xv[0];
        const float4 b = xv[1];
        const float v[NQ] = {a.x, a.y, a.z, a.w, b.x, b.y, b.z, b.w};
        const float q = scale * row_prod(v);
        *reinterpret_cast<float2*>(out + (size_t)2 * row) = make_float2(q, -q);
    }
#endif
}

extern "C" void kernel_launch(void* const* d_in, const int* in_sizes, int n_in,
                              void* d_out, int out_size, void* d_ws, size_t ws_size,
                              hipStream_t stream) {
    (void)in_sizes; (void)n_in; (void)out_size; (void)d_ws; (void)ws_size;
    const float* x     = (const float*)d_in[0];
    // d_in[1] = weights: unused by the closed-form circuit (n_layers = 0)
    const float* scale = (const float*)d_in[2];
    float* out         = (float*)d_out;

    qnn_expz_kernel<<<dim3(GRID), dim3(BLOCK), 0, stream>>>(x, scale, out);
}